// EntityNamePredictor_4406636445857
// MI455X (gfx1250) — compile-verified
//
#include <hip/hip_runtime.h>
#include <hip/hip_bf16.h>

#define BSZ  64
#define SEQ  4096
#define DIM  400
#define NENT 32

typedef __attribute__((ext_vector_type(2))) float v2f;
typedef __attribute__((ext_vector_type(8))) float v8f;

// ---------------------------------------------------------------------------
// Hardware f32 atomic add (no return -> STOREcnt; S_ENDPGM waits idle).
// ---------------------------------------------------------------------------
__device__ __forceinline__ void atomic_add_f32(float* p, float v) {
    asm volatile("global_atomic_add_f32 %0, %1, off"
                 :
                 : "v"(p), "v"(v)
                 : "memory");
}

// ---------------------------------------------------------------------------
// Kernel 1: per-batch-row scan of is_b -> slot index per position (-1 invalid)
// One block (256 threads) per batch row; each thread owns 16 consecutive
// positions; Hillis-Steele scan in LDS across the 256 partial counts.
// ---------------------------------------------------------------------------
__global__ void scan_kernel(const int* __restrict__ labels,
                            int* __restrict__ slot,
                            float* __restrict__ num_slots_out) {
    __shared__ int sums[256];
    const int b = blockIdx.x;
    const int t = threadIdx.x;
    const int base = b * SEQ + t * 16;

    int lab[16];
    int cnt = 0;
#pragma unroll
    for (int i = 0; i < 16; ++i) {
        lab[i] = labels[base + i];
        cnt += (lab[i] == 1) ? 1 : 0;
    }
    sums[t] = cnt;
    __syncthreads();

    for (int off = 1; off < 256; off <<= 1) {
        int v = (t >= off) ? sums[t - off] : 0;
        __syncthreads();
        sums[t] += v;
        __syncthreads();
    }

    int seg = (t > 0) ? sums[t - 1] : 0;  // exclusive prefix of is_b count
#pragma unroll
    for (int i = 0; i < 16; ++i) {
        seg += (lab[i] == 1) ? 1 : 0;
        const bool valid = ((lab[i] == 1) || (lab[i] == 2)) && (seg > 0);
        slot[base + i] = valid ? (seg - 1) : -1;
    }
    if (t == 255) num_slots_out[b] = (float)sums[255];
}

// ---------------------------------------------------------------------------
// Kernel 2: zero the logits region of d_out (harness poisons it).
// ---------------------------------------------------------------------------
__global__ void zero_kernel(float4* __restrict__ out, long n4) {
    const float4 z = make_float4(0.f, 0.f, 0.f, 0.f);
    for (long i = (long)blockIdx.x * blockDim.x + threadIdx.x;
         i < n4; i += (long)gridDim.x * blockDim.x)
        out[i] = z;
}

// ---------------------------------------------------------------------------
// Kernel 3: WMMA projection + atomic segment scatter.
// One wave per 16-seq-position tile. A = 16x4 tile of hidden (f32, NT loads),
// B = 4x16 tile of E (f32, from LDS), C/D = 16x16 f32 accumulator.
// K-loop: DIM/4 = 100 iterations; two N-tiles cover the 32 entity types.
//
// ISA VGPR layouts (CDNA5 ISA):
//   A 16x4:  lane L holds row M = L&15; VGPR0 -> K = 2*(L>>4), VGPR1 -> K+1
//   B 4x16:  lane L holds col N = L&15; VGPR0 -> K = 2*(L>>4), VGPR1 -> K+1
//   D 16x16: VGPR j, lane L -> D[j + 8*(L>>4), L&15]
//
// E is staged in LDS K-pair-interleaved: eLds[(k/2)*NENT + n] = (E[k][n],
// E[k+1][n]) so every B operand is ONE aligned ds_load_b64 straight into a
// consecutive VGPR pair (no v_mov stitching in the hot loop).
// ---------------------------------------------------------------------------
__global__ void __launch_bounds__(256)
wmma_scatter_kernel(const float* __restrict__ H,
                    const float* __restrict__ E,
                    const int* __restrict__ slot,
                    float* __restrict__ logits) {
    __shared__ v2f eLds[(DIM / 2) * NENT];  // 51.2 KB of the 320 KB/WGP LDS
    for (int i = threadIdx.x; i < (DIM / 2) * NENT; i += blockDim.x) {
        const int p = i / NENT;       // K-pair index (k = 2p, 2p+1)
        const int c = i - p * NENT;   // entity column
        v2f e;
        e.x = E[(2 * p) * NENT + c];
        e.y = E[(2 * p + 1) * NENT + c];
        eLds[i] = e;
    }
    __syncthreads();

    const int lane = threadIdx.x & 31;
    const int wave = threadIdx.x >> 5;
    const int tile = blockIdx.x * 8 + wave;  // 16384 tiles total
    const int b    = tile >> 8;              // 256 tiles per batch row
    const int s0   = (tile & 255) << 4;
    const int hi   = lane >> 4;
    const int lo   = lane & 15;

    const float* hrow = H + ((size_t)b * SEQ + s0 + lo) * DIM;

    v8f c0 = {};
    v8f c1 = {};
    for (int kk = 0; kk < DIM; kk += 4) {
        const int ka = kk + 2 * hi;          // even; this lane's K base
        const int p  = ka >> 1;              // K-pair index
        // hidden is streamed once (419 MB): non-temporal so it does not
        // evict E / logits (33.5 MB) from the 192 MB L2.
        const v2f a  = __builtin_nontemporal_load((const v2f*)(hrow + ka));
        const v2f b0 = eLds[p * NENT + lo];
        const v2f b1 = eLds[p * NENT + 16 + lo];
        c0 = __builtin_amdgcn_wmma_f32_16x16x4_f32(false, a, false, b0,
                                                   (short)0, c0, false, false);
        c1 = __builtin_amdgcn_wmma_f32_16x16x4_f32(false, a, false, b1,
                                                   (short)0, c1, false, false);
    }

    // Scatter: row j+8*hi of the tile -> logits[b, slot, lo] / [.., 16+lo].
    // This lane needs the 8 slot indices of rows 8*hi..8*hi+7: fetch them
    // with two b128 loads up front (srow+8*hi is 16B-aligned) so we pay one
    // load-wait instead of eight serialized L2 round-trips.
    const int4* sv = (const int4*)(slot + b * SEQ + s0 + 8 * hi);
    const int4  sA = sv[0];
    const int4  sB = sv[1];
    int s8[8] = { sA.x, sA.y, sA.z, sA.w, sB.x, sB.y, sB.z, sB.w };

    float* lbase = logits + (size_t)b * SEQ * NENT;
#pragma unroll
    for (int j = 0; j < 8; ++j) {
        const int s = s8[j];
        if (s >= 0) {
            float* p = lbase + (size_t)s * NENT + lo;
            atomic_add_f32(p, c0[j]);
            atomic_add_f32(p + 16, c1[j]);
        }
    }
}

// ---------------------------------------------------------------------------
extern "C" void kernel_launch(void* const* d_in, const int* in_sizes, int n_in,
                              void* d_out, int out_size, void* d_ws, size_t ws_size,
                              hipStream_t stream) {
    const float* H      = (const float*)d_in[0];  // (64, 4096, 400) f32
    const int*   labels = (const int*)d_in[1];    // (64, 4096) i32
    const float* E      = (const float*)d_in[2];  // (400, 32) f32

    float* logits    = (float*)d_out;                       // (64, 4096, 32)
    float* num_slots = logits + (size_t)BSZ * SEQ * NENT;   // (64,)
    int*   slot      = (int*)d_ws;                          // 1 MB scratch

    scan_kernel<<<BSZ, 256, 0, stream>>>(labels, slot, num_slots);

    const long n4 = (long)BSZ * SEQ * NENT / 4;
    zero_kernel<<<2048, 256, 0, stream>>>((float4*)logits, n4);

    // 16384 tiles of 16 seq positions, 8 waves per block
    wmma_scatter_kernel<<<2048, 256, 0, stream>>>(H, E, slot, logits);
}